// ApproxLayerNorm_3719441678849
// MI455X (gfx1250) — compile-verified
//
#include <hip/hip_runtime.h>
#include <stdint.h>

// ApproxLayerNorm, MI455X/gfx1250.
// Bandwidth-bound (256 MB traffic -> ~11us floor @ 23.3 TB/s). Exact integer
// fixed-point path (floor-divs are power-of-two shifts). Uses CDNA5
// global_load_async_to_lds_b128 + s_wait_asynccnt to stage weight/bias into
// LDS overlapped with the statistics pass; wave32 shuffle reductions.

#define N_SEG   32
#define DVAL    4096
#define NCHUNK  16      // G
#define TPB     256     // 8 wave32s
#define NJ      4       // float4 iterations per thread
#define VPT     16      // values per thread

__device__ __forceinline__ int wave_reduce_i32(int v) {
#pragma unroll
  for (int off = 1; off < 32; off <<= 1) v += __shfl_xor(v, off, 32);
  return v;
}

__device__ __forceinline__ long long wave_reduce_i64(long long v) {
  unsigned long long u = (unsigned long long)v;
#pragma unroll
  for (int off = 1; off < 32; off <<= 1) {
    unsigned lo  = (unsigned)u;
    unsigned hi  = (unsigned)(u >> 32);
    unsigned lo2 = (unsigned)__shfl_xor((int)lo, off, 32);
    unsigned hi2 = (unsigned)__shfl_xor((int)hi, off, 32);
    u += (((unsigned long long)hi2) << 32) | (unsigned long long)lo2;
  }
  return (long long)u;
}

// Replicates jnp pwl_approx: idx = clip(searchsorted(breaks, v, 'right')-1, 0, N-1),
// out = slopes[idx]*v + intercepts[idx], with v < breaks[0] -> 0.
__device__ __forceinline__ float pwl_eval(float v,
                                          const float* __restrict__ breaks,
                                          const float* __restrict__ slopes,
                                          const float* __restrict__ intercepts) {
  if (v < breaks[0]) return 0.0f;
  int idx = -1;
  for (int i = 0; i < N_SEG + 1; ++i) idx += (v >= breaks[i]) ? 1 : 0;
  if (idx > N_SEG - 1) idx = N_SEG - 1;
  return slopes[idx] * v + intercepts[idx];
}

__global__ __launch_bounds__(TPB) void approx_layernorm_q88_kernel(
    const float* __restrict__ x, const float* __restrict__ weight,
    const float* __restrict__ bias,
    const float* __restrict__ sb, const float* __restrict__ ss, const float* __restrict__ si,
    const float* __restrict__ rb, const float* __restrict__ rs, const float* __restrict__ ri,
    float* __restrict__ out) {
  __shared__ float lds_w[DVAL];
  __shared__ float lds_b[DVAL];
  __shared__ int lds_sum[NCHUNK];
  __shared__ unsigned long long lds_M[NCHUNK];
  __shared__ long long lds_mu;
  __shared__ float lds_scale;

  const int t    = threadIdx.x;
  const int lane = t & 31;
  const int row  = blockIdx.x;

  // ---- Phase 0: kick off async DMA of weight/bias into LDS (CDNA5 path). ----
  // Overlapped with the entire statistics phase; fenced by s_wait_asynccnt.
  {
    const char* wptr = (const char*)weight;
    const char* bptr = (const char*)bias;
    unsigned lw = (unsigned)(uintptr_t)(&lds_w[0]);
    unsigned lb = (unsigned)(uintptr_t)(&lds_b[0]);
#pragma unroll
    for (int j = 0; j < NJ; ++j) {
      unsigned byteOff = (unsigned)(t + j * TPB) * 16u;
      asm volatile("global_load_async_to_lds_b128 %0, %1, off"
                   :: "v"(lw + byteOff), "v"(wptr + byteOff) : "memory");
      asm volatile("global_load_async_to_lds_b128 %0, %1, off"
                   :: "v"(lb + byteOff), "v"(bptr + byteOff) : "memory");
    }
  }

  if (t < NCHUNK) { lds_sum[t] = 0; lds_M[t] = 0ull; }

  // ---- Phase 1: load row (b128), quantize to q8.8 codes, per-chunk partials.
  // Element e = j*1024 + t*4 + k  ->  chunk c = 4*j + (t>>6); whole wave32 is
  // inside one chunk for each j, so a wave butterfly + one LDS atomic suffices.
  const float4* xrow = (const float4*)(x + (size_t)row * DVAL);
  int xq[VPT];
  int part[NJ];
#pragma unroll
  for (int j = 0; j < NJ; ++j) {
    float4 v = xrow[t + j * TPB];
    int q0 = __float2int_rn(v.x * 256.0f);   // jnp.round = RNE
    int q1 = __float2int_rn(v.y * 256.0f);
    int q2 = __float2int_rn(v.z * 256.0f);
    int q3 = __float2int_rn(v.w * 256.0f);
    xq[4 * j + 0] = q0; xq[4 * j + 1] = q1;
    xq[4 * j + 2] = q2; xq[4 * j + 3] = q3;
    part[j] = q0 + q1 + q2 + q3;
  }

  __syncthreads();  // lds_sum/lds_M init visible

#pragma unroll
  for (int j = 0; j < NJ; ++j) {
    int s = wave_reduce_i32(part[j]);
    if (lane == 0) atomicAdd(&lds_sum[4 * j + (t >> 6)], s);
  }
  __syncthreads();

  // ---- Phase 2: per-chunk sum of squared deviations (exact int64). ----
#pragma unroll
  for (int j = 0; j < NJ; ++j) {
    int c  = 4 * j + (t >> 6);
    int mu = lds_sum[c] >> 8;                // floor-div by 256 (arith shift)
    long long sq = 0;
#pragma unroll
    for (int k = 0; k < 4; ++k) {
      int d = xq[4 * j + k] - mu;
      sq += (long long)d * (long long)d;
    }
    sq = wave_reduce_i64(sq);
    if (lane == 0) atomicAdd(&lds_M[c], (unsigned long long)sq);
  }
  __syncthreads();

  // ---- Phase 3: thread 0 does the exact pairwise tree merge + PWLs. ----
  if (t == 0) {
    long long mu[NCHUNK], M[NCHUNK], total = 0;
#pragma unroll
    for (int c = 0; c < NCHUNK; ++c) {
      int s = lds_sum[c];
      total += (long long)s;
      mu[c] = ((long long)s) >> 8;           // chunk mean, floor-div 256
      M[c]  = (long long)lds_M[c];
    }
    long long mu_q = total >> 12;            // row mean, floor-div 4096

    long long ncur = 256;
    int g = NCHUNK;
    while (g > 1) {
      for (int i = 0; i < g / 2; ++i) {
        long long m1 = mu[2 * i], m2 = mu[2 * i + 1];
        long long dd = m1 - m2;
        // (d^2 * n * n) // (2n) == (d^2 * n) >> 1  (non-negative)
        M[i]  = M[2 * i] + M[2 * i + 1] + ((dd * dd * ncur) >> 1);
        mu[i] = (m1 + m2) >> 1;              // floor-div 2 (arith shift)
      }
      ncur <<= 1;
      g >>= 1;
    }
    long long var_q16 = M[0] >> 12;          // // D
    float var = (float)(int)(var_q16 >> 8) / 256.0f;  // q8.8 -> float

    float sqrt_var = pwl_eval(var + 1e-5f, sb, ss, si);
    float inv_sqrt = pwl_eval(sqrt_var, rb, rs, ri);
    lds_mu    = mu_q;
    lds_scale = inv_sqrt;
  }

  // Each wave fences its own async LDS fills, then barrier makes all visible.
  asm volatile("s_wait_asynccnt 0x0" ::: "memory");
  __syncthreads();

  // ---- Phase 4: normalize + affine, b128 stores; w/b served from LDS. ----
  const long long mu_q = lds_mu;
  const float scale = lds_scale;
  float4* orow = (float4*)(out + (size_t)row * DVAL);
#pragma unroll
  for (int j = 0; j < NJ; ++j) {
    int e0 = j * (TPB * 4) + t * 4;
    float4 o;
    {
      float xc = (float)((long long)xq[4 * j + 0] - mu_q) / 256.0f;
      o.x = (xc * scale) * lds_w[e0 + 0] + lds_b[e0 + 0];
    }
    {
      float xc = (float)((long long)xq[4 * j + 1] - mu_q) / 256.0f;
      o.y = (xc * scale) * lds_w[e0 + 1] + lds_b[e0 + 1];
    }
    {
      float xc = (float)((long long)xq[4 * j + 2] - mu_q) / 256.0f;
      o.z = (xc * scale) * lds_w[e0 + 2] + lds_b[e0 + 2];
    }
    {
      float xc = (float)((long long)xq[4 * j + 3] - mu_q) / 256.0f;
      o.w = (xc * scale) * lds_w[e0 + 3] + lds_b[e0 + 3];
    }
    orow[t + j * TPB] = o;
  }
}

extern "C" void kernel_launch(void* const* d_in, const int* in_sizes, int n_in,
                              void* d_out, int out_size, void* d_ws, size_t ws_size,
                              hipStream_t stream) {
  (void)n_in; (void)d_ws; (void)ws_size; (void)out_size;
  const float* x  = (const float*)d_in[0];
  const float* w  = (const float*)d_in[1];
  const float* b  = (const float*)d_in[2];
  const float* sb = (const float*)d_in[3];
  const float* ss = (const float*)d_in[4];
  const float* si = (const float*)d_in[5];
  const float* rb = (const float*)d_in[6];
  const float* rs = (const float*)d_in[7];
  const float* ri = (const float*)d_in[8];
  float* out = (float*)d_out;

  const int D = in_sizes[1];            // 4096
  const int B = in_sizes[0] / D;        // 8192
  approx_layernorm_q88_kernel<<<B, TPB, 0, stream>>>(x, w, b, sb, ss, si,
                                                     rb, rs, ri, out);
}